// RMultiHeadAttentionLayer_4080218931585
// MI455X (gfx1250) — compile-verified
//
#include <hip/hip_runtime.h>
#include <hip/hip_bf16.h>
#include <math.h>

typedef __attribute__((ext_vector_type(2))) float v2f;
typedef __attribute__((ext_vector_type(8))) float v8f;

#define DD    128
#define CDIM  384      // concat(Q|K|V) columns
#define NH    8
#define DHD   16
#define RREL  6
#define BBAS  6
#define LPAD  132      // padded LDS row stride (floats): bank-conflict free

// ---------------------------------------------------------------------------
// Build Wall[r][i][c] = sum_b coef[r,b] * basis[b,i,o], c = o for Q / o+128 K / o+256 V
// ---------------------------------------------------------------------------
__global__ void build_w_kernel(const float* __restrict__ qB, const float* __restrict__ qC,
                               const float* __restrict__ kB, const float* __restrict__ kC,
                               const float* __restrict__ vB, const float* __restrict__ vC,
                               float* __restrict__ Wall) {
    int idx = blockIdx.x * blockDim.x + threadIdx.x;
    if (idx >= RREL * DD * CDIM) return;
    int c = idx % CDIM;
    int i = (idx / CDIM) % DD;
    int r = idx / (CDIM * DD);
    const float* basis; const float* coef; int o;
    if (c < 128)      { basis = qB; coef = qC; o = c; }
    else if (c < 256) { basis = kB; coef = kC; o = c - 128; }
    else              { basis = vB; coef = vC; o = c - 256; }
    float s = 0.f;
#pragma unroll
    for (int b = 0; b < BBAS; ++b)
        s += coef[r * BBAS + b] * basis[(b * DD + i) * DD + o];
    Wall[idx] = s;
}

// ---------------------------------------------------------------------------
// WMMA fp32 GEMM: tmp[N,384] = h[N,128] @ W[128,384]   (one relation)
// grid = (N/16, 3), block = 256 (8 waves). Wave w handles cols [by*128+w*16, +16).
// ---------------------------------------------------------------------------
__global__ __launch_bounds__(256) void gemm_qkv_kernel(const float* __restrict__ h,
                                                       const float* __restrict__ W,
                                                       float* __restrict__ out) {
    __shared__ float lh[16 * LPAD];
    const int row0 = blockIdx.x * 16;
    const int t = threadIdx.x;
    // stage 16x128 tile of h into LDS (coalesced)
#pragma unroll
    for (int i = 0; i < 8; ++i) {
        int flat = t + 256 * i;                 // 0..2047
        int rr = flat >> 7, cc = flat & 127;
        lh[rr * LPAD + cc] = h[(size_t)(row0 + rr) * DD + cc];
    }
    __syncthreads();

    const int wave = t >> 5, lane = t & 31;
    const int g = lane >> 4;        // lane group 0/1
    const int m = lane & 15;        // row (A) / col (B,C)
    const int col0 = blockIdx.y * 128 + wave * 16;
    const float* Wc = W + col0 + m;

    v8f c = {};
#pragma unroll
    for (int kk = 0; kk < DD; kk += 4) {
        int ka = kk + 2 * g;
        v2f a, b;
        a.x = lh[m * LPAD + ka];
        a.y = lh[m * LPAD + ka + 1];
        b.x = Wc[(size_t)ka * CDIM];
        b.y = Wc[(size_t)(ka + 1) * CDIM];
        c = __builtin_amdgcn_wmma_f32_16x16x4_f32(false, a, false, b,
                                                  (short)0, c, false, false);
    }
    float* orow = out + col0 + m;
#pragma unroll
    for (int v = 0; v < 8; ++v)
        orow[(size_t)(row0 + v + 8 * g) * CDIM] = c[v];
}

// ---------------------------------------------------------------------------
// Per-relation edge scatter: agg[dst] += tmp[src] for etype==rel. One wave/edge.
// ---------------------------------------------------------------------------
__global__ __launch_bounds__(256) void scatter_rel_kernel(const float* __restrict__ tmp,
                                                          const int* __restrict__ src,
                                                          const int* __restrict__ dst,
                                                          const int* __restrict__ et,
                                                          float* __restrict__ agg,
                                                          int rel, int E) {
    int e = (blockIdx.x * blockDim.x + threadIdx.x) >> 5;
    int lane = threadIdx.x & 31;
    if (e >= E) return;
    if (et[e] != rel) return;
    const float4* srow = (const float4*)(tmp + (size_t)src[e] * CDIM);
    float* drow = agg + (size_t)dst[e] * CDIM;
#pragma unroll
    for (int i = 0; i < 3; ++i) {
        int c4 = lane + 32 * i;                 // 96 float4 = 384 floats
        float4 v = srow[c4];
        unsafeAtomicAdd(drow + c4 * 4 + 0, v.x);
        unsafeAtomicAdd(drow + c4 * 4 + 1, v.y);
        unsafeAtomicAdd(drow + c4 * 4 + 2, v.z);
        unsafeAtomicAdd(drow + c4 * 4 + 3, v.w);
    }
}

// ---------------------------------------------------------------------------
// qkv = relu(agg + bias)
// ---------------------------------------------------------------------------
__global__ void bias_relu_kernel(const float* __restrict__ agg,
                                 const float* __restrict__ bq,
                                 const float* __restrict__ bk,
                                 const float* __restrict__ bv,
                                 float* __restrict__ qkv, int total) {
    int i = blockIdx.x * blockDim.x + threadIdx.x;
    if (i >= total) return;
    int c = i % CDIM;
    float b = (c < 128) ? bq[c] : (c < 256) ? bk[c - 128] : bv[c - 256];
    float x = agg[i] + b;
    qkv[i] = x > 0.f ? x : 0.f;
}

// ---------------------------------------------------------------------------
// Attention edge pass. One wave/edge: 4-lane groups per head, shfl_xor reduce.
// ---------------------------------------------------------------------------
__global__ __launch_bounds__(256) void attn_edge_kernel(const float* __restrict__ qkv,
                                                        const int* __restrict__ src,
                                                        const int* __restrict__ dst,
                                                        float* __restrict__ out,
                                                        float* __restrict__ z, int E) {
    int e = (blockIdx.x * blockDim.x + threadIdx.x) >> 5;
    int lane = threadIdx.x & 31;
    if (e >= E) return;
    int s = src[e], d = dst[e];
    const float4* Qd = (const float4*)(qkv + (size_t)d * CDIM);        // Q[dst]
    const float4* Ks = (const float4*)(qkv + (size_t)s * CDIM + 128);  // K[src]
    const float4* Vs = (const float4*)(qkv + (size_t)s * CDIM + 256);  // V[src]
    float4 q = Qd[lane], k = Ks[lane], v = Vs[lane];
    float p = q.x * k.x + q.y * k.y + q.z * k.z + q.w * k.w;   // partial head dot
    p += __shfl_xor(p, 1, 32);
    p += __shfl_xor(p, 2, 32);                                  // full dot per 4-lane head group
    float sc = p * 0.25f;                                       // / sqrt(DH=16)
    sc = fminf(10.f, fmaxf(-10.f, sc));
    sc = expf(sc);
    float* orow = out + (size_t)d * DD;
    int c0 = lane * 4;
    unsafeAtomicAdd(orow + c0 + 0, v.x * sc);
    unsafeAtomicAdd(orow + c0 + 1, v.y * sc);
    unsafeAtomicAdd(orow + c0 + 2, v.z * sc);
    unsafeAtomicAdd(orow + c0 + 3, v.w * sc);
    if ((lane & 3) == 0)
        unsafeAtomicAdd(z + (size_t)d * NH + (lane >> 2), sc);
}

// ---------------------------------------------------------------------------
// out = wV / (z + 1e-6)
// ---------------------------------------------------------------------------
__global__ void finalize_kernel(float* __restrict__ out, const float* __restrict__ z,
                                int total) {
    int i = blockIdx.x * blockDim.x + threadIdx.x;
    if (i >= total) return;
    int n = i >> 7, c = i & 127;
    out[i] = out[i] / (z[(size_t)n * NH + (c >> 4)] + 1e-6f);
}

// ---------------------------------------------------------------------------
extern "C" void kernel_launch(void* const* d_in, const int* in_sizes, int n_in,
                              void* d_out, int out_size, void* d_ws, size_t ws_size,
                              hipStream_t stream) {
    const float* h   = (const float*)d_in[0];
    const int*   src = (const int*)d_in[1];
    const int*   dst = (const int*)d_in[2];
    const int*   et  = (const int*)d_in[3];
    const float* WqB = (const float*)d_in[4];
    const float* Wqc = (const float*)d_in[5];
    const float* bq  = (const float*)d_in[6];
    const float* WkB = (const float*)d_in[7];
    const float* Wkc = (const float*)d_in[8];
    const float* bk  = (const float*)d_in[9];
    const float* WvB = (const float*)d_in[10];
    const float* Wvc = (const float*)d_in[11];
    const float* bv  = (const float*)d_in[12];

    const int N = in_sizes[0] / DD;   // 50000 (divisible by 16)
    const int E = in_sizes[1];        // 500000
    float* out = (float*)d_out;

    // workspace layout
    char* ws = (char*)d_ws;
    float* Wall = (float*)ws;                                   // 6*128*384 f32
    size_t off = ((size_t)RREL * DD * CDIM * 4 + 255) & ~(size_t)255;
    float* tmp = (float*)(ws + off); off += ((size_t)N * CDIM * 4 + 255) & ~(size_t)255;
    float* agg = (float*)(ws + off); off += ((size_t)N * CDIM * 4 + 255) & ~(size_t)255;
    float* z   = (float*)(ws + off);

    hipMemsetAsync(agg, 0, (size_t)N * CDIM * 4, stream);
    hipMemsetAsync(z,   0, (size_t)N * NH * 4, stream);
    hipMemsetAsync(out, 0, (size_t)N * DD * 4, stream);

    build_w_kernel<<<(RREL * DD * CDIM + 255) / 256, 256, 0, stream>>>(
        WqB, Wqc, WkB, Wkc, WvB, Wvc, Wall);

    dim3 ggrid(N / 16, 3);
    int eblocks = (E * 32 + 255) / 256;   // one wave per edge
    for (int r = 0; r < RREL; ++r) {
        gemm_qkv_kernel<<<ggrid, 256, 0, stream>>>(h, Wall + (size_t)r * DD * CDIM, tmp);
        scatter_rel_kernel<<<eblocks, 256, 0, stream>>>(tmp, src, dst, et, agg, r, E);
    }

    int tot = N * CDIM;
    bias_relu_kernel<<<(tot + 255) / 256, 256, 0, stream>>>(agg, bq, bk, bv, tmp, tot);
    attn_edge_kernel<<<eblocks, 256, 0, stream>>>(tmp, src, dst, out, z, E);
    finalize_kernel<<<(N * DD + 255) / 256, 256, 0, stream>>>(out, z, N * DD);
}